// MHGAN_9242769621445
// MI455X (gfx1250) — compile-verified
//
#include <hip/hip_runtime.h>
#include <math.h>

#define NNODES 100000
#define EDGES  500000
#define MP     2
#define DIM    64
#define HID    128
#define BATCH  8192

typedef float v2f __attribute__((ext_vector_type(2)));
typedef float v8f __attribute__((ext_vector_type(8)));

__device__ __forceinline__ float lrelu(float x){ return x > 0.f ? x : 0.2f * x; }

// float atomic max via sign-split integer trick (valid for mixed signs; init -inf)
__device__ __forceinline__ void atomicMaxFloat(float* addr, float val){
    if (val >= 0.f) atomicMax((int*)addr, __float_as_int(val));
    else            atomicMin((unsigned int*)addr, (unsigned int)__float_as_int(val));
}

__global__ void fill_kernel(float* __restrict__ p, float v, int n){
    int i = blockIdx.x * blockDim.x + threadIdx.x;
    if (i < n) p[i] = v;
}

// ---------------------------------------------------------------------------
// C[N x DOUT] = A[N x 64] @ B[64 x DOUT], fp32 WMMA 16x16x4.
// One 16-row tile per block; each wave owns one 16-col tile (blockDim = 32*DOUT/16).
// A-frag layout (ISA 7.12.2): lane L -> M = L%16 ; VGPR k of pair -> K = k0 + 2*(L>=16) + k
// C/D layout: VGPR r -> M = r + 8*(L>=16), N = L%16.
// ---------------------------------------------------------------------------
template<int DOUT>
__global__ void gemm_k64_wmma(const float* __restrict__ A, const float* __restrict__ B,
                              float* __restrict__ C){
    const int lane    = threadIdx.x & 31;
    const int wave    = threadIdx.x >> 5;
    const int col     = wave * 16 + (lane & 15);
    const int half    = lane >> 4;                 // 0 -> K{0,1}, 1 -> K{2,3}
    const int rowBase = blockIdx.x * 16;
    const int arow    = rowBase + (lane & 15);
    v8f c = {};
    #pragma unroll
    for (int k0 = 0; k0 < 64; k0 += 4){
        const int k = k0 + 2 * half;
        v2f a, b;
        a.x = A[(size_t)arow * 64 + k];
        a.y = A[(size_t)arow * 64 + k + 1];
        b.x = B[(size_t)k * DOUT + col];
        b.y = B[(size_t)(k + 1) * DOUT + col];
        c = __builtin_amdgcn_wmma_f32_16x16x4_f32(false, a, false, b, (short)0, c, false, false);
    }
    #pragma unroll
    for (int r = 0; r < 8; ++r)
        C[(size_t)(rowBase + r + 8 * half) * DOUT + col] = c[r];
}

// ---------------------------------------------------------------------------
// Semantic-attention GEMM, fused epilogue:
//   wacc += sum_rows( tanh(Z @ W1 + b1) @ W2 )      (mean's 1/N applied later)
// Z: (N,64), W1: (64,128), b1: (128), W2: (128). blockDim = 256 (8 waves).
// ---------------------------------------------------------------------------
__global__ void sem_gemm_wmma(const float* __restrict__ Z, const float* __restrict__ W1,
                              const float* __restrict__ b1, const float* __restrict__ W2,
                              float* __restrict__ wacc){
    const int lane    = threadIdx.x & 31;
    const int wave    = threadIdx.x >> 5;
    const int col     = wave * 16 + (lane & 15);
    const int half    = lane >> 4;
    const int arow    = blockIdx.x * 16 + (lane & 15);
    v8f c = {};
    #pragma unroll
    for (int k0 = 0; k0 < 64; k0 += 4){
        const int k = k0 + 2 * half;
        v2f a, b;
        a.x = Z[(size_t)arow * 64 + k];
        a.y = Z[(size_t)arow * 64 + k + 1];
        b.x = W1[(size_t)k * HID + col];
        b.y = W1[(size_t)(k + 1) * HID + col];
        c = __builtin_amdgcn_wmma_f32_16x16x4_f32(false, a, false, b, (short)0, c, false, false);
    }
    const float bb = b1[col], w2 = W2[col];
    float partial = 0.f;
    #pragma unroll
    for (int r = 0; r < 8; ++r) partial += tanhf(c[r] + bb) * w2;
    #pragma unroll
    for (int off = 16; off > 0; off >>= 1) partial += __shfl_down(partial, off, 32);
    if (lane == 0) atomicAdd(wacc, partial);
}

// el[n] = feat[n,:] . al ; er[n] = feat[n,:] . ar   (one wave per node)
__global__ void attn_logits_kernel(const float* __restrict__ feat, const float* __restrict__ al,
                                   const float* __restrict__ ar, float* __restrict__ el,
                                   float* __restrict__ er, int n){
    const int lane = threadIdx.x & 31;
    const int node = blockIdx.x * (blockDim.x >> 5) + (threadIdx.x >> 5);
    if (node >= n) return;
    const float f0 = feat[(size_t)node * 64 + lane];
    const float f1 = feat[(size_t)node * 64 + lane + 32];
    float a = f0 * al[lane] + f1 * al[lane + 32];
    float b = f0 * ar[lane] + f1 * ar[lane + 32];
    #pragma unroll
    for (int off = 16; off > 0; off >>= 1){
        a += __shfl_down(a, off, 32);
        b += __shfl_down(b, off, 32);
    }
    if (lane == 0){ el[node] = a; er[node] = b; }
}

__global__ void edge_max_kernel(const int* __restrict__ src, const int* __restrict__ dst,
                                const float* __restrict__ el, const float* __restrict__ er,
                                float* __restrict__ mx){
    int e = blockIdx.x * blockDim.x + threadIdx.x;
    if (e >= EDGES) return;
    int s = src[e], d = dst[e];
    atomicMaxFloat(&mx[d], lrelu(el[s] + er[d]));
}

__global__ void edge_sumexp_kernel(const int* __restrict__ src, const int* __restrict__ dst,
                                   const float* __restrict__ el, const float* __restrict__ er,
                                   const float* __restrict__ mx, float* __restrict__ den){
    int e = blockIdx.x * blockDim.x + threadIdx.x;
    if (e >= EDGES) return;
    int s = src[e], d = dst[e];
    float x = lrelu(el[s] + er[d]);
    atomicAdd(&den[d], expf(x - mx[d]));
}

// one wave per edge: z[d,:] += alpha * feat[s,:]
__global__ void edge_scatter_kernel(const int* __restrict__ src, const int* __restrict__ dst,
                                    const float* __restrict__ el, const float* __restrict__ er,
                                    const float* __restrict__ mx, const float* __restrict__ den,
                                    const float* __restrict__ feat, float* __restrict__ zout){
    const int lane = threadIdx.x & 31;
    const int e = blockIdx.x * (blockDim.x >> 5) + (threadIdx.x >> 5);
    if (e >= EDGES) return;
    const int s = src[e], d = dst[e];
    const float x = lrelu(el[s] + er[d]);
    const float alpha = expf(x - mx[d]) / (den[d] + 1e-9f);
    atomicAdd(&zout[(size_t)d * 64 + lane],      alpha * feat[(size_t)s * 64 + lane]);
    atomicAdd(&zout[(size_t)d * 64 + lane + 32], alpha * feat[(size_t)s * 64 + lane + 32]);
}

__global__ void elu_bias_kernel(float* __restrict__ z, const float* __restrict__ b, int n64){
    int i = blockIdx.x * blockDim.x + threadIdx.x;
    if (i >= n64) return;
    float v = z[i] + b[i & 63];
    z[i] = v > 0.f ? v : (expf(v) - 1.f);
}

// beta = softmax over metapaths of (wacc / N); wacc layout [u0,u1,i0,i1]
__global__ void beta_kernel(const float* __restrict__ wacc, float* __restrict__ beta){
    if (threadIdx.x != 0 || blockIdx.x != 0) return;
    for (int t = 0; t < 2; ++t){
        float w0 = wacc[2 * t]     * (1.f / (float)NNODES);
        float w1 = wacc[2 * t + 1] * (1.f / (float)NNODES);
        float m  = fmaxf(w0, w1);
        float e0 = expf(w0 - m), e1 = expf(w1 - m);
        float inv = 1.f / (e0 + e1);
        beta[2 * t] = e0 * inv; beta[2 * t + 1] = e1 * inv;
    }
}

// Gathered rows only: combine beta, project (64x64), ReLU, LayerNorm. Wave per row.
__global__ void final_kernel(const int* __restrict__ uidx, const int* __restrict__ iidx,
                             const int* __restrict__ nidx,
                             const float* __restrict__ zu, const float* __restrict__ zi,
                             const float* __restrict__ beta,
                             const float* __restrict__ uW, const float* __restrict__ ub,
                             const float* __restrict__ iW, const float* __restrict__ ib,
                             const float* __restrict__ g, const float* __restrict__ bln,
                             float* __restrict__ out){
    const int lane = threadIdx.x & 31;
    const int row = blockIdx.x * (blockDim.x >> 5) + (threadIdx.x >> 5);
    if (row >= 3 * BATCH) return;
    const int seg = row / BATCH, i = row % BATCH;
    int idx; const float *z, *bt, *W, *bias;
    if (seg == 0){ idx = uidx[i]; z = zu; bt = beta;     W = uW; bias = ub; }
    else { idx = (seg == 1 ? iidx[i] : nidx[i]); z = zi; bt = beta + 2; W = iW; bias = ib; }
    const float b0 = bt[0], b1v = bt[1];
    const float* z0 = z + (size_t)idx * 64;
    const float* z1 = z + (size_t)NNODES * 64 + (size_t)idx * 64;
    const float x0 = b0 * z0[lane]      + b1v * z1[lane];
    const float x1 = b0 * z0[lane + 32] + b1v * z1[lane + 32];
    float y0 = 0.f, y1 = 0.f;
    for (int k = 0; k < 32; ++k){
        float xk = __shfl(x0, k, 32);
        y0 += xk * W[k * 64 + lane];
        y1 += xk * W[k * 64 + lane + 32];
    }
    for (int k = 0; k < 32; ++k){
        float xk = __shfl(x1, k, 32);
        y0 += xk * W[(k + 32) * 64 + lane];
        y1 += xk * W[(k + 32) * 64 + lane + 32];
    }
    y0 = fmaxf(y0 + bias[lane], 0.f);
    y1 = fmaxf(y1 + bias[lane + 32], 0.f);
    float s = y0 + y1;
    #pragma unroll
    for (int off = 16; off > 0; off >>= 1) s += __shfl_xor(s, off, 32);
    const float mu = s * (1.f / 64.f);
    const float d0 = y0 - mu, d1 = y1 - mu;
    float v = d0 * d0 + d1 * d1;
    #pragma unroll
    for (int off = 16; off > 0; off >>= 1) v += __shfl_xor(v, off, 32);
    const float rstd = rsqrtf(v * (1.f / 64.f) + 1e-5f);
    out[(size_t)row * 64 + lane]      = g[lane]      * d0 * rstd + bln[lane];
    out[(size_t)row * 64 + lane + 32] = g[lane + 32] * d1 * rstd + bln[lane + 32];
}

extern "C" void kernel_launch(void* const* d_in, const int* in_sizes, int n_in,
                              void* d_out, int out_size, void* d_ws, size_t ws_size,
                              hipStream_t stream){
    const int*   user_idx  = (const int*)  d_in[0];
    const int*   item_idx  = (const int*)  d_in[1];
    const int*   neg_idx   = (const int*)  d_in[2];
    const float* user_feat = (const float*)d_in[3];
    const float* item_feat = (const float*)d_in[4];
    const int*   u_src     = (const int*)  d_in[5];
    const int*   u_dst     = (const int*)  d_in[6];
    const int*   i_src     = (const int*)  d_in[7];
    const int*   i_dst     = (const int*)  d_in[8];
    const float* u_W  = (const float*)d_in[9];
    const float* u_al = (const float*)d_in[10];
    const float* u_ar = (const float*)d_in[11];
    const float* u_b  = (const float*)d_in[12];
    const float* i_W  = (const float*)d_in[13];
    const float* i_al = (const float*)d_in[14];
    const float* i_ar = (const float*)d_in[15];
    const float* i_b  = (const float*)d_in[16];
    const float* u_saW1 = (const float*)d_in[17];
    const float* u_sab1 = (const float*)d_in[18];
    const float* u_saW2 = (const float*)d_in[19];
    const float* i_saW1 = (const float*)d_in[20];
    const float* i_sab1 = (const float*)d_in[21];
    const float* i_saW2 = (const float*)d_in[22];
    const float* userW  = (const float*)d_in[23];
    const float* userb  = (const float*)d_in[24];
    const float* itemW  = (const float*)d_in[25];
    const float* itemb  = (const float*)d_in[26];
    const float* ln_g   = (const float*)d_in[27];
    const float* ln_b   = (const float*)d_in[28];
    (void)in_sizes; (void)n_in; (void)out_size; (void)ws_size;

    // workspace layout (floats): ~130 MB total
    float* ws = (float*)d_ws;
    const size_t ND = (size_t)NNODES * 64;
    const size_t NZ = (size_t)MP * ND;
    float* z_u     = ws;                 // 12.8M
    float* z_i     = z_u + NZ;           // 12.8M
    float* featbuf = z_i + NZ;           // 6.4M
    float* el      = featbuf + ND;       // 100k
    float* er      = el  + NNODES;
    float* mx      = er  + NNODES;
    float* den     = mx  + NNODES;
    float* wacc    = den + NNODES;       // 4
    float* beta    = wacc + 4;           // 4

    auto fill = [&](float* p, float v, size_t n){
        fill_kernel<<<(unsigned)((n + 255) / 256), 256, 0, stream>>>(p, v, (int)n);
    };

    fill(z_u, 0.f, NZ);
    fill(z_i, 0.f, NZ);
    fill(wacc, 0.f, 4);

    const int rowTiles = NNODES / 16;                     // 6250 (exact)
    const int edgeBlk  = (EDGES + 255) / 256;
    const int edgeWaveBlk = (EDGES + 7) / 8;              // 8 waves/block

    for (int t = 0; t < 2; ++t){
        const float* hfeat = t ? item_feat : user_feat;
        const int*   srcA  = t ? i_src : u_src;
        const int*   dstA  = t ? i_dst : u_dst;
        const float* Wm    = t ? i_W  : u_W;
        const float* alm   = t ? i_al : u_al;
        const float* arm   = t ? i_ar : u_ar;
        const float* bm    = t ? i_b  : u_b;
        const float* saW1  = t ? i_saW1 : u_saW1;
        const float* sab1  = t ? i_sab1 : u_sab1;
        const float* saW2  = t ? i_saW2 : u_saW2;
        float*       z     = t ? z_i : z_u;

        for (int m = 0; m < MP; ++m){
            gemm_k64_wmma<64><<<rowTiles, 128, 0, stream>>>(hfeat, Wm + (size_t)m * 64 * 64, featbuf);
            attn_logits_kernel<<<(NNODES + 7) / 8, 256, 0, stream>>>(
                featbuf, alm + m * 64, arm + m * 64, el, er, NNODES);
            fill(mx, -INFINITY, NNODES);
            fill(den, 0.f, NNODES);
            edge_max_kernel<<<edgeBlk, 256, 0, stream>>>(
                srcA + (size_t)m * EDGES, dstA + (size_t)m * EDGES, el, er, mx);
            edge_sumexp_kernel<<<edgeBlk, 256, 0, stream>>>(
                srcA + (size_t)m * EDGES, dstA + (size_t)m * EDGES, el, er, mx, den);
            edge_scatter_kernel<<<edgeWaveBlk, 256, 0, stream>>>(
                srcA + (size_t)m * EDGES, dstA + (size_t)m * EDGES, el, er, mx, den,
                featbuf, z + (size_t)m * ND);
            elu_bias_kernel<<<(int)((ND + 255) / 256), 256, 0, stream>>>(
                z + (size_t)m * ND, bm + m * 64, (int)ND);
            sem_gemm_wmma<<<rowTiles, 256, 0, stream>>>(
                z + (size_t)m * ND, saW1, sab1, saW2, wacc + 2 * t + m);
        }
    }

    beta_kernel<<<1, 32, 0, stream>>>(wacc, beta);
    final_kernel<<<(3 * BATCH + 7) / 8, 256, 0, stream>>>(
        user_idx, item_idx, neg_idx, z_u, z_i, beta,
        userW, userb, itemW, itemb, ln_g, ln_b, (float*)d_out);
}